// VoxCPM2DecodeStepWrapper_7395933684330
// MI455X (gfx1250) — compile-verified
//
#include <hip/hip_runtime.h>
#include <hip/hip_bf16.h>
#include <math.h>

// ---- model dims ----
#define Lc     16
#define Bc     4
#define Dc     1024
#define Hc     16
#define HDc    64
#define KVHc   8
#define DFFc   4096
#define Sc     2048
#define SP1c   2049
#define Gc     (Hc / KVHc)          // 2 query heads per KV head
#define EPSc   1e-5f
#define THETAc 10000.0f
#define MUPc   0.35f                // 1.4 / sqrt(16)
#define ASCALE 0.125f               // 1 / sqrt(64)

typedef __attribute__((ext_vector_type(2))) float v2f;
typedef __attribute__((ext_vector_type(8))) float v8f;
typedef __attribute__((ext_vector_type(4))) unsigned int u32x4;
typedef __attribute__((ext_vector_type(8))) int i32x8;
typedef __attribute__((ext_vector_type(4))) int i32x4;

// ------------------------------------------------------------------
// TDM tile fetch: 2D tile (16 rows x 64 cols, fp32, row stride = N)
// from global into LDS.  D# layout per CDNA5 ISA 8.3/8.4: group0 =
// {count/type, lds_addr, global_addr}; group1 = data_size, tensor
// dims (huge - tile always in-bounds), tile dims, row stride.
// Groups 2/3 zero (2D tensor).  Tracked on TENSORcnt.
// ------------------------------------------------------------------
#define TDM_ROWS 16
#define TDM_COLS 64

__device__ __forceinline__ void tdm_load_tile(const float* gptr,
                                              unsigned lds_off, int nstride) {
  const unsigned long long ga = (unsigned long long)(uintptr_t)gptr;
  u32x4 g0;
  g0.x = 1u;                                          // count=1, user mode
  g0.y = lds_off;                                     // lds_addr (bytes)
  g0.z = (unsigned)(ga & 0xffffffffu);                // global_addr[31:0]
  g0.w = (unsigned)((ga >> 32) & 0x01ffffffu) | (2u << 30);  // [56:32] | type=2
  const unsigned td0 = 1u << 30, td1 = 1u << 30;      // huge tensor dims
  const unsigned long long st0 = (unsigned long long)(unsigned)nstride;
  i32x8 g1;
  g1[0] = (int)(2u << 16);                            // wg_mask=0, data_size=4B
  g1[1] = (int)((td0 & 0xffffu) << 16);               // barrier_addr=0 | td0.lo
  g1[2] = (int)((td0 >> 16) | ((td1 & 0xffffu) << 16));
  g1[3] = (int)((td1 >> 16) | ((unsigned)TDM_COLS << 16));   // tile_dim0=64
  g1[4] = (int)TDM_ROWS;                              // tile_dim1=16, tile_dim2=0
  g1[5] = (int)(st0 & 0xffffffffu);                   // tensor_dim0_stride lo
  g1[6] = (int)((st0 >> 32) & 0xffffu);               // stride hi | dim1_stride=0
  g1[7] = 0;
  const i32x4 z4 = {0, 0, 0, 0};
  const i32x8 z8 = {0, 0, 0, 0, 0, 0, 0, 0};
#if __has_builtin(__builtin_amdgcn_tensor_load_to_lds)
  __builtin_amdgcn_tensor_load_to_lds(g0, g1, z4, z4, z8, 0);
#endif
}

// ------------------------------------------------------------------
// TDM-staged WMMA GEMV for all weight projections:
//   out[B, N] (+)= x[B, K] @ W[K, N]
// Double-buffered tensor_load_to_lds -> s_wait_tensorcnt -> ds_load
// B-fragments -> V_WMMA_F32_16X16X4_F32.  One block = 4 waves = one
// 64-column tile over a split-K chunk; tiny outputs merged with fp32
// global atomics.  A-frag per ISA 16x4 f32 layout: lanes 0-15 carry
// K = k0,k0+1; lanes 16-31 carry K = k0+2,k0+3; M = lane&15 with
// rows >= B zero-padded (compute is free, HBM streaming is the bound).
// ------------------------------------------------------------------
__global__ void gemv_wmma_tdm_kernel(const float* __restrict__ x,
                                     const float* __restrict__ W,
                                     float* __restrict__ out,
                                     int K, int N, int kchunk) {
  __shared__ float smem[2][TDM_ROWS * TDM_COLS];
  const int wave  = threadIdx.x >> 5;
  const int lane  = threadIdx.x & 31;
  const int ncol0 = blockIdx.x * TDM_COLS;
  const int subcol = wave * 16 + (lane & 15);
  const int m     = lane & 15;
  const int kofs  = (lane >> 4) * 2;
  const int k0beg = blockIdx.y * kchunk;
  const int nsteps = kchunk / TDM_ROWS;
  const unsigned lds0 = (unsigned)(uintptr_t)&smem[0][0];
  const unsigned lds1 = (unsigned)(uintptr_t)&smem[1][0];

  if (wave == 0)
    tdm_load_tile(W + (long)k0beg * N + ncol0, lds0, N);

  v8f c = {};
  for (int s = 0; s < nsteps; ++s) {
    const int k0 = k0beg + s * TDM_ROWS;
    if (wave == 0) {
      if (s + 1 < nsteps) {
        tdm_load_tile(W + (long)(k0 + TDM_ROWS) * N + ncol0,
                      (s & 1) ? lds0 : lds1, N);
        __builtin_amdgcn_s_wait_tensorcnt((short)1);   // current tile done
      } else {
        __builtin_amdgcn_s_wait_tensorcnt((short)0);   // last tile done
      }
    }
    __syncthreads();
    const float* buf = (s & 1) ? smem[1] : smem[0];
    #pragma unroll
    for (int kk = 0; kk < TDM_ROWS; kk += 4) {
      const int ka = kk + kofs;
      v2f a, b;
      a.x = (m < Bc) ? x[m * K + k0 + ka]     : 0.0f;
      a.y = (m < Bc) ? x[m * K + k0 + ka + 1] : 0.0f;
      b.x = buf[ka * TDM_COLS + subcol];
      b.y = buf[(ka + 1) * TDM_COLS + subcol];
      c = __builtin_amdgcn_wmma_f32_16x16x4_f32(false, a, false, b,
                                                (short)0, c, false, false);
    }
    __syncthreads();                                   // before buffer reuse
  }
  if (lane < 16) {
    #pragma unroll
    for (int r = 0; r < Bc; ++r)
      atomicAdd(&out[r * N + ncol0 + wave * 16 + lane], c[r]);
  }
}

// ------------------------------------------------------------------
__global__ void rmsnorm_kernel(const float* __restrict__ in,
                               const float* __restrict__ w,
                               float* __restrict__ out) {
  __shared__ float red[256];
  const int row = blockIdx.x;
  const int t = threadIdx.x;
  float s = 0.f;
  for (int i = t; i < Dc; i += 256) { float v = in[row * Dc + i]; s += v * v; }
  red[t] = s; __syncthreads();
  for (int o = 128; o > 0; o >>= 1) { if (t < o) red[t] += red[t + o]; __syncthreads(); }
  const float r = rsqrtf(red[0] / (float)Dc + EPSc);
  for (int i = t; i < Dc; i += 256) out[row * Dc + i] = in[row * Dc + i] * r * w[i];
}

__global__ void rope_q_kernel(float* __restrict__ q, const int* __restrict__ posp) {
  const int head = blockIdx.x;              // b*H + h
  const int d = threadIdx.x;
  const int j = d & 31;
  const float pos = (float)(*posp);
  const float fr = pos * __powf(THETAc, -(float)(2 * j) / (float)HDc);
  const float cs = __cosf(fr), sn = __sinf(fr);
  const float v0 = q[head * HDc + d];
  const float vo = q[head * HDc + ((d < 32) ? d + 32 : d - 32)];
  __syncthreads();
  const float rot = (d < 32) ? -vo : vo;
  q[head * HDc + d] = v0 * cs + rot * sn;
}

__global__ void rope_append_kv_kernel(const float* __restrict__ k,
                                      const float* __restrict__ v,
                                      float* __restrict__ nk,
                                      float* __restrict__ nv,
                                      int layer, const int* __restrict__ posp) {
  const int bk = blockIdx.x;                // b*KVH + kvh
  const int d = threadIdx.x;
  const int j = d & 31;
  const float pos = (float)(*posp);
  const float fr = pos * __powf(THETAc, -(float)(2 * j) / (float)HDc);
  const float cs = __cosf(fr), sn = __sinf(fr);
  const float k0 = k[bk * HDc + d];
  const float ko = k[bk * HDc + ((d < 32) ? d + 32 : d - 32)];
  const float rot = (d < 32) ? -ko : ko;
  const long rowbase = (((long)layer * Bc * KVHc + bk) * SP1c + Sc) * HDc;
  nk[rowbase + d] = k0 * cs + rot * sn;
  nv[rowbase + d] = v[bk * HDc + d];
}

__global__ void attn_kernel(const float* __restrict__ q,
                            const float* __restrict__ nk,
                            const float* __restrict__ nv,
                            float* __restrict__ o, int layer) {
  __shared__ float sc[Gc][SP1c];
  __shared__ float red[256];
  __shared__ float inv[Gc];
  const int bk = blockIdx.x;                // b*KVH + kvh
  const int t = threadIdx.x, lane = t & 31, wave = t >> 5;
  const long kbase = ((long)layer * Bc * KVHc + bk) * (long)SP1c * HDc;
  const int qbase = bk * Gc * HDc;
  const float q0a = q[qbase + lane],       q0b = q[qbase + 32 + lane];
  const float q1a = q[qbase + HDc + lane], q1b = q[qbase + HDc + 32 + lane];
  for (int s = wave; s < SP1c; s += 8) {
    const float ka = nk[kbase + (long)s * HDc + lane];
    const float kb = nk[kbase + (long)s * HDc + 32 + lane];
    float p0 = q0a * ka + q0b * kb;
    float p1 = q1a * ka + q1b * kb;
    for (int off = 16; off; off >>= 1) {
      p0 += __shfl_xor(p0, off);
      p1 += __shfl_xor(p1, off);
    }
    if (lane == 0) { sc[0][s] = p0 * ASCALE; sc[1][s] = p1 * ASCALE; }
  }
  __syncthreads();
  for (int g = 0; g < Gc; ++g) {
    float m = -INFINITY;
    for (int s = t; s < SP1c; s += 256) m = fmaxf(m, sc[g][s]);
    red[t] = m; __syncthreads();
    for (int o2 = 128; o2; o2 >>= 1) { if (t < o2) red[t] = fmaxf(red[t], red[t + o2]); __syncthreads(); }
    m = red[0]; __syncthreads();
    float lsum = 0.f;
    for (int s = t; s < SP1c; s += 256) { float e = __expf(sc[g][s] - m); sc[g][s] = e; lsum += e; }
    red[t] = lsum; __syncthreads();
    for (int o2 = 128; o2; o2 >>= 1) { if (t < o2) red[t] += red[t + o2]; __syncthreads(); }
    if (t == 0) inv[g] = 1.f / red[0];
    __syncthreads();
  }
  if (t < HDc) {
    float a0 = 0.f, a1 = 0.f;
    for (int s = 0; s < SP1c; ++s) {
      const float vv = nv[kbase + (long)s * HDc + t];
      a0 += sc[0][s] * vv;
      a1 += sc[1][s] * vv;
    }
    o[qbase + t]       = a0 * inv[0];
    o[qbase + HDc + t] = a1 * inv[1];
  }
}

__global__ void silu_mul_kernel(const float* __restrict__ g,
                                const float* __restrict__ u,
                                float* __restrict__ a, int n) {
  const int i = blockIdx.x * blockDim.x + threadIdx.x;
  if (i < n) { const float x = g[i]; a[i] = (x / (1.f + __expf(-x))) * u[i]; }
}

__global__ void axpy_kernel(float* __restrict__ h, const float* __restrict__ a,
                            float alpha, int n) {
  const int i = blockIdx.x * blockDim.x + threadIdx.x;
  if (i < n) h[i] += a[i] * alpha;
}

__global__ void copy_cache_kernel(const float4* __restrict__ kc,
                                  const float4* __restrict__ vc,
                                  float4* __restrict__ nk,
                                  float4* __restrict__ nv) {
  long idx = (long)blockIdx.x * blockDim.x + threadIdx.x;
  const long f4row = HDc / 4;                            // 16
  const long total = (long)Lc * Bc * KVHc * Sc * f4row;  // per cache
  if (idx >= 2 * total) return;
  const float4* src = kc; float4* dst = nk;
  if (idx >= total) { idx -= total; src = vc; dst = nv; }
  const long row = idx / f4row, c = idx % f4row;
  const long bh = row / Sc, s = row % Sc;
  dst[(bh * SP1c + s) * f4row + c] = src[row * f4row + c];
}

__global__ void tail_i64_kernel(const int* __restrict__ cl, long long* __restrict__ out) {
  if (threadIdx.x == 0 && blockIdx.x == 0) *out = (long long)(*cl) + 1;
}
__global__ void tail_i32_kernel(const int* __restrict__ cl, int* __restrict__ out) {
  if (threadIdx.x == 0 && blockIdx.x == 0) *out = *cl + 1;
}

// ------------------------------------------------------------------
extern "C" void kernel_launch(void* const* d_in, const int* in_sizes, int n_in,
                              void* d_out, int out_size, void* d_ws, size_t ws_size,
                              hipStream_t stream) {
  const float* emb = (const float*)d_in[0];
  const float* kc  = (const float*)d_in[1];
  const float* vc  = (const float*)d_in[2];
  const float* ln1 = (const float*)d_in[3];
  const float* ln2 = (const float*)d_in[4];
  const float* nw  = (const float*)d_in[5];
  const float* wq  = (const float*)d_in[6];
  const float* wk  = (const float*)d_in[7];
  const float* wv  = (const float*)d_in[8];
  const float* wo  = (const float*)d_in[9];
  const float* wg  = (const float*)d_in[10];
  const float* wu  = (const float*)d_in[11];
  const float* wd  = (const float*)d_in[12];
  const int*   cl  = (const int*)d_in[13];

  float* out = (float*)d_out;
  const long cacheElems = (long)Lc * Bc * KVHc * SP1c * HDc;   // 67,141,632
  float* h_out = out;
  float* nk = out + (long)Bc * Dc;
  float* nv = nk + cacheElems;
  float* tailp = nv + cacheElems;

  float* W    = (float*)d_ws;
  float* h    = W;                 // 4096
  float* x    = W + 4096;          // 4096
  float* qb   = W + 8192;          // 4096
  float* kb   = W + 12288;         // 2048
  float* vb   = W + 14336;         // 2048
  float* ob   = W + 16384;         // 4096
  float* tmp  = W + 20480;         // 4096
  float* gb   = W + 24576;         // 16384
  float* ub   = W + 40960;         // 16384

  (void)hipMemcpyAsync(h, emb, sizeof(float) * Bc * Dc, hipMemcpyDeviceToDevice, stream);

  {
    const long tot = 2L * Lc * Bc * KVHc * Sc * (HDc / 4);
    copy_cache_kernel<<<dim3((unsigned)((tot + 255) / 256)), 256, 0, stream>>>(
        (const float4*)kc, (const float4*)vc, (float4*)nk, (float4*)nv);
  }

  const int KSPLIT = 8;
  for (int l = 0; l < Lc; ++l) {
    const float* Wq = wq + (long)l * Dc * Hc * HDc;
    const float* Wk = wk + (long)l * Dc * KVHc * HDc;
    const float* Wv = wv + (long)l * Dc * KVHc * HDc;
    const float* Wo = wo + (long)l * Hc * HDc * Dc;
    const float* Wg = wg + (long)l * Dc * DFFc;
    const float* Wu = wu + (long)l * Dc * DFFc;
    const float* Wd = wd + (long)l * DFFc * Dc;

    // ---- attention ----
    rmsnorm_kernel<<<Bc, 256, 0, stream>>>(h, ln1 + l * Dc, x);
    (void)hipMemsetAsync(qb, 0, sizeof(float) * (4096 + 2048 + 2048), stream);
    gemv_wmma_tdm_kernel<<<dim3((Hc * HDc) / TDM_COLS, KSPLIT), 128, 0, stream>>>(
        x, Wq, qb, Dc, Hc * HDc, Dc / KSPLIT);
    gemv_wmma_tdm_kernel<<<dim3((KVHc * HDc) / TDM_COLS, KSPLIT), 128, 0, stream>>>(
        x, Wk, kb, Dc, KVHc * HDc, Dc / KSPLIT);
    gemv_wmma_tdm_kernel<<<dim3((KVHc * HDc) / TDM_COLS, KSPLIT), 128, 0, stream>>>(
        x, Wv, vb, Dc, KVHc * HDc, Dc / KSPLIT);
    rope_q_kernel<<<Bc * Hc, 64, 0, stream>>>(qb, cl);
    rope_append_kv_kernel<<<Bc * KVHc, 64, 0, stream>>>(kb, vb, nk, nv, l, cl);
    attn_kernel<<<Bc * KVHc, 256, 0, stream>>>(qb, nk, nv, ob, l);
    (void)hipMemsetAsync(tmp, 0, sizeof(float) * Bc * Dc, stream);
    gemv_wmma_tdm_kernel<<<dim3(Dc / TDM_COLS, KSPLIT), 128, 0, stream>>>(
        ob, Wo, tmp, Hc * HDc, Dc, (Hc * HDc) / KSPLIT);
    axpy_kernel<<<16, 256, 0, stream>>>(h, tmp, MUPc, Bc * Dc);

    // ---- MLP (SwiGLU) ----
    rmsnorm_kernel<<<Bc, 256, 0, stream>>>(h, ln2 + l * Dc, x);
    (void)hipMemsetAsync(gb, 0, sizeof(float) * 2 * Bc * DFFc, stream);
    gemv_wmma_tdm_kernel<<<dim3(DFFc / TDM_COLS, KSPLIT), 128, 0, stream>>>(
        x, Wg, gb, Dc, DFFc, Dc / KSPLIT);
    gemv_wmma_tdm_kernel<<<dim3(DFFc / TDM_COLS, KSPLIT), 128, 0, stream>>>(
        x, Wu, ub, Dc, DFFc, Dc / KSPLIT);
    silu_mul_kernel<<<(Bc * DFFc + 255) / 256, 256, 0, stream>>>(gb, ub, gb, Bc * DFFc);
    (void)hipMemsetAsync(tmp, 0, sizeof(float) * Bc * Dc, stream);
    gemv_wmma_tdm_kernel<<<dim3(Dc / TDM_COLS, KSPLIT), 128, 0, stream>>>(
        gb, Wd, tmp, DFFc, Dc, DFFc / KSPLIT);
    axpy_kernel<<<16, 256, 0, stream>>>(h, tmp, MUPc, Bc * Dc);
  }

  rmsnorm_kernel<<<Bc, 256, 0, stream>>>(h, nw, h_out);

  const long used = (long)Bc * Dc + 2 * cacheElems;
  if ((long)out_size - used >= 2)
    tail_i64_kernel<<<1, 1, 0, stream>>>(cl, (long long*)tailp);
  else
    tail_i32_kernel<<<1, 1, 0, stream>>>(cl, (int*)tailp);
}